// PVEncoder_44770739093811
// MI455X (gfx1250) — compile-verified
//
#include <hip/hip_runtime.h>
#include <hip/hip_bf16.h>

typedef float v2f __attribute__((ext_vector_type(2)));
typedef float v4f __attribute__((ext_vector_type(4)));
typedef float v8f __attribute__((ext_vector_type(8)));

#define DGRID 16
#define HGRID 256
#define WGRID 256
#define NVOX  40000
#define NPTS  200000
#define SSLOT 20

// ---------------------------------------------------------------- utilities
__global__ void zero_f32(float* __restrict__ p, long n) {
  long i = (long)blockIdx.x * blockDim.x + threadIdx.x;
  long stride = (long)gridDim.x * blockDim.x;
  for (; i < n; i += stride) p[i] = 0.f;
}

__global__ void init_vox(float* __restrict__ h, int* __restrict__ counts,
                         int* __restrict__ minidx) {
  int i = blockIdx.x * blockDim.x + threadIdx.x;
  if (i < NVOX * 32) h[i] = 0.f;
  if (i < NVOX) { counts[i] = 0; minidx[i] = 0x7fffffff; }
}

// ------------------------------------------------- point -> voxel encoder
__global__ void point_accum(const float* __restrict__ points,
                            const float* __restrict__ voxels,
                            const float* __restrict__ w_in,
                            const float* __restrict__ s_in,
                            const float* __restrict__ t_in,
                            const int* __restrict__ p2v,
                            float* __restrict__ h,
                            int* __restrict__ counts,
                            int* __restrict__ minidx) {
  int i = blockIdx.x * blockDim.x + threadIdx.x;
  if (i >= NPTS) return;
  int v = p2v[i];
  atomicAdd(&counts[v], 1);
  atomicMin(&minidx[v], i);
  float f[8];
#pragma unroll
  for (int c = 0; c < 4; c++) {
    float pc = points[(long)c * NPTS + i];
    f[c] = voxels[v * 4 + c] - pc;
    f[4 + c] = pc;
  }
#pragma unroll
  for (int o = 0; o < 32; o++) {
    float d = 0.f;
#pragma unroll
    for (int c = 0; c < 8; c++) d += w_in[o * 8 + c] * f[c];
    float g = fmaxf(d * s_in[o] + t_in[o], 0.f);
    atomicAdd(&h[v * 32 + o], g);
  }
}

__global__ void voxel_finalize(const float* __restrict__ points,
                               const float* __restrict__ voxels,
                               const float* __restrict__ w_in,
                               const float* __restrict__ s_in,
                               const float* __restrict__ t_in,
                               const int* __restrict__ counts,
                               const int* __restrict__ minidx,
                               float* __restrict__ h) {
  int v = blockIdx.x * blockDim.x + threadIdx.x;
  if (v >= NVOX) return;
  int cnt = counts[v];
  const float inv = 1.f / (float)SSLOT;
  if (cnt <= 0) {
#pragma unroll
    for (int o = 0; o < 32; o++) h[v * 32 + o] = 0.f;
    return;
  }
  int i = minidx[v];
  float f[8];
#pragma unroll
  for (int c = 0; c < 4; c++) {
    float pc = points[(long)c * NPTS + i];
    f[c] = voxels[v * 4 + c] - pc;
    f[4 + c] = pc;
  }
  int cc = cnt < SSLOT ? cnt : SSLOT;
  float rep = (float)(SSLOT - cc);
#pragma unroll
  for (int o = 0; o < 32; o++) {
    float d = 0.f;
#pragma unroll
    for (int c = 0; c < 8; c++) d += w_in[o * 8 + c] * f[c];
    float g = fmaxf(d * s_in[o] + t_in[o], 0.f);
    h[v * 32 + o] = (h[v * 32 + o] + rep * g) * inv;
  }
}

// --------------------------------------------------------------- scatter
// x0p is zero-padded channels-last: [18][258][258][32]
__global__ void scatter_vox(const int* __restrict__ coors,
                            const float* __restrict__ h,
                            float* __restrict__ x0p,
                            float* __restrict__ occ0) {
  int v = blockIdx.x * blockDim.x + threadIdx.x;
  if (v >= NVOX) return;
  int xc = coors[v * 4 + 1];
  int yc = coors[v * 4 + 2];
  int zc = coors[v * 4 + 3];
  occ0[((long)zc * HGRID + yc) * WGRID + xc] = 1.f;
  long sp = (((long)(zc + 1) * (HGRID + 2) + (yc + 1)) * (WGRID + 2) + (xc + 1)) * 32;
#pragma unroll
  for (int c = 0; c < 32; c++) x0p[sp + c] = h[v * 32 + c];
}

// ------------------------------------------------------------- pool mask
__global__ void pool_mask(const float* __restrict__ in, float* __restrict__ out,
                          int Di, int Hi, int Wi, int Do, int Ho, int Wo,
                          int stride) {
  long i = (long)blockIdx.x * blockDim.x + threadIdx.x;
  long tot = (long)Do * Ho * Wo;
  if (i >= tot) return;
  int x = (int)(i % Wo);
  int y = (int)((i / Wo) % Ho);
  int z = (int)(i / ((long)Wo * Ho));
  float m = 0.f;
#pragma unroll
  for (int kd = 0; kd < 3; kd++) {
    int zi = z * stride + kd - 1;
    if (zi < 0 || zi >= Di) continue;
#pragma unroll
    for (int kh = 0; kh < 3; kh++) {
      int yi = y * stride + kh - 1;
      if (yi < 0 || yi >= Hi) continue;
#pragma unroll
      for (int kw = 0; kw < 3; kw++) {
        int xi = x * stride + kw - 1;
        if (xi < 0 || xi >= Wi) continue;
        m = fmaxf(m, in[((long)zi * Hi + yi) * Wi + xi]);
      }
    }
  }
  out[i] = m;
}

// ----------------------------------------------- implicit-GEMM conv (WMMA)
// Input: zero-padded channels-last [Din+2][Hin+2][Win+2][CIN] -> branchless.
// One wave: 16 couts x (NT*16) W-positions. A staged in LDS as float2 pairs.
// CLAST=true: write padded channels-last output; false: write NCDHW (d_out).
template <int CIN, int COUT, int STRIDE, int NT, bool CLAST>
__global__ __launch_bounds__(256) void conv3_wmma(
    const float* __restrict__ in, const float* __restrict__ w,
    const float* __restrict__ s, const float* __restrict__ t,
    const float* __restrict__ mask, float* __restrict__ out, int inHp,
    int inWp, int Dout, int Hout, int Wout, int outHp, int outWp) {
  // ---- stage 16 x CIN x 27 weight tile in LDS, packed as ci-pairs:
  // lw[((ci>>1)*27 + tap)*16 + m] = {w[ci], w[ci+1]}  (conflict-free b64 reads)
  constexpr int WTOT = 16 * CIN * 27;
  __shared__ float lw[WTOT];
  const int coT = blockIdx.y;
  {
    const float* gw = w + (size_t)coT * 16 * CIN * 27;
    for (int g = threadIdx.x; g < WTOT; g += 256) {
      int mm = g / (CIN * 27);
      int r = g % (CIN * 27);
      int ci = r / 27;
      int tap = r % 27;
      lw[((((ci >> 1) * 27) + tap) * 16 + mm) * 2 + (ci & 1)] = gw[g];
    }
  }
  __syncthreads();

  const int lane = threadIdx.x & 31;
  const int sw = blockIdx.x * (blockDim.x >> 5) + (threadIdx.x >> 5);
  const int nWG = Wout / (16 * NT);
  int r = sw;
  const int wg = r % nWG;  r /= nWG;
  const int yo = r % Hout; r /= Hout;
  const int zo = r;
  if (zo >= Dout) return;  // wave-uniform

  const int n = lane & 15;   // output position within tile / A row (cout)
  const int kp = lane >> 4;  // K-pair select {0,1} vs {2,3}

  int wadd[NT];
#pragma unroll
  for (int j = 0; j < NT; j++)
    wadd[j] = ((wg * (16 * NT) + j * 16 + n) * STRIDE) * CIN;

  v8f acc[NT];
#pragma unroll
  for (int j = 0; j < NT; j++) acc[j] = (v8f){0.f, 0.f, 0.f, 0.f, 0.f, 0.f, 0.f, 0.f};

  const v2f* __restrict__ lwp = (const v2f*)lw;

#pragma unroll
  for (int kd = 0; kd < 3; kd++) {
#pragma unroll
    for (int kh = 0; kh < 3; kh++) {
      const int rowbase = ((zo * STRIDE + kd) * inHp + (yo * STRIDE + kh)) * inWp;
#pragma unroll
      for (int kw = 0; kw < 3; kw++) {
        const int tap = (kd * 3 + kh) * 3 + kw;
        const float* __restrict__ tp = in + (rowbase + kw) * CIN;
        for (int ci0 = 0; ci0 < CIN; ci0 += 4) {
          const int cpair = (ci0 >> 1) + kp;          // ci = ci0 + 2*kp
          const v2f a = lwp[(cpair * 27 + tap) * 16 + n];
#pragma unroll
          for (int j = 0; j < NT; j++) {
            const v2f b = *(const v2f*)(tp + wadd[j] + ci0 + 2 * kp);
            acc[j] = __builtin_amdgcn_wmma_f32_16x16x4_f32(
                false, a, false, b, (short)0, acc[j], false, false);
          }
        }
      }
    }
  }

  // ---- epilogue: BN + ReLU + occupancy mask
#pragma unroll
  for (int j = 0; j < NT; j++) {
    const int wo = wg * (16 * NT) + j * 16 + n;
    const float mk = mask[((long)zo * Hout + yo) * Wout + wo];
    float vv[8];
#pragma unroll
    for (int rr = 0; rr < 8; rr++) {
      const int co = coT * 16 + kp * 8 + rr;
      vv[rr] = fmaxf(acc[j][rr] * s[co] + t[co], 0.f) * mk;
    }
    if (CLAST) {
      float* p = out +
                 (((long)(zo + 1) * outHp + (yo + 1)) * outWp + (wo + 1)) * COUT +
                 coT * 16 + kp * 8;
      *(v4f*)p = (v4f){vv[0], vv[1], vv[2], vv[3]};
      *(v4f*)(p + 4) = (v4f){vv[4], vv[5], vv[6], vv[7]};
    } else {
#pragma unroll
      for (int rr = 0; rr < 8; rr++) {
        const int co = coT * 16 + kp * 8 + rr;
        out[((long)co * Dout + zo) * (long)Hout * Wout + (long)yo * Wout + wo] =
            vv[rr];
      }
    }
  }
}

// ---------------------------------------------------------------- launcher
static inline int divup(long a, long b) { return (int)((a + b - 1) / b); }

extern "C" void kernel_launch(void* const* d_in, const int* in_sizes, int n_in,
                              void* d_out, int out_size, void* d_ws,
                              size_t ws_size, hipStream_t stream) {
  (void)in_sizes; (void)n_in; (void)out_size; (void)ws_size;
  const float* points = (const float*)d_in[0];
  const float* voxels = (const float*)d_in[1];
  const float* w_in = (const float*)d_in[2];
  const float* s_in = (const float*)d_in[3];
  const float* t_in = (const float*)d_in[4];
  const float *W[10], *SS[10], *TT[10];
  for (int i = 0; i < 9; i++) {
    W[i] = (const float*)d_in[5 + 3 * i];
    SS[i] = (const float*)d_in[6 + 3 * i];
    TT[i] = (const float*)d_in[7 + 3 * i];
  }
  W[9] = (const float*)d_in[32];
  SS[9] = (const float*)d_in[33];
  TT[9] = (const float*)d_in[34];
  const int* coors = (const int*)d_in[35];
  const int* p2v = (const int*)d_in[36];
  float* out = (float*)d_out;

  // workspace carve (256B aligned)
  char* wp = (char*)d_ws;
  auto carve = [&](size_t bytes) -> void* {
    void* r = (void*)wp;
    wp += (bytes + 255) & ~(size_t)255;
    return r;
  };
  float* h = (float*)carve((size_t)NVOX * 32 * 4);
  int* counts = (int*)carve((size_t)NVOX * 4);
  int* minidx = (int*)carve((size_t)NVOX * 4);
  float* occ0 = (float*)carve((size_t)DGRID * HGRID * WGRID * 4);
  float* occ1 = (float*)carve((size_t)8 * 128 * 128 * 4);
  float* occ2 = (float*)carve((size_t)4 * 64 * 64 * 4);
  float* occ3 = (float*)carve((size_t)2 * 32 * 32 * 4);
  float* occ4 = (float*)carve((size_t)2 * 32 * 32 * 4);
  // padded channels-last activations (contiguous region -> single bulk zero)
  char* padBegin = wp;
  float* x0p = (float*)carve((size_t)18 * 258 * 258 * 32 * 4);
  float* s0A = (float*)carve((size_t)10 * 130 * 130 * 32 * 4);
  float* s0B = (float*)carve((size_t)10 * 130 * 130 * 32 * 4);
  float* s1A = (float*)carve((size_t)6 * 66 * 66 * 64 * 4);
  float* s1B = (float*)carve((size_t)6 * 66 * 66 * 64 * 4);
  float* s2A = (float*)carve((size_t)4 * 34 * 34 * 64 * 4);
  float* s2B = (float*)carve((size_t)4 * 34 * 34 * 64 * 4);
  const long padFloats = (long)(wp - padBegin) / 4;

  // ---- point->voxel encoder
  init_vox<<<divup(NVOX * 32, 256), 256, 0, stream>>>(h, counts, minidx);
  point_accum<<<divup(NPTS, 256), 256, 0, stream>>>(points, voxels, w_in, s_in,
                                                    t_in, p2v, h, counts,
                                                    minidx);
  voxel_finalize<<<divup(NVOX, 256), 256, 0, stream>>>(points, voxels, w_in,
                                                       s_in, t_in, counts,
                                                       minidx, h);

  // ---- zero padded activations + occ0, then dense scatter
  zero_f32<<<16384, 256, 0, stream>>>((float*)padBegin, padFloats);
  zero_f32<<<2048, 256, 0, stream>>>(occ0, (long)DGRID * HGRID * WGRID);
  scatter_vox<<<divup(NVOX, 256), 256, 0, stream>>>(coors, h, x0p, occ0);

  // ---- occupancy pyramid
  pool_mask<<<divup((long)8 * 128 * 128, 256), 256, 0, stream>>>(
      occ0, occ1, 16, 256, 256, 8, 128, 128, 2);
  pool_mask<<<divup((long)4 * 64 * 64, 256), 256, 0, stream>>>(
      occ1, occ2, 8, 128, 128, 4, 64, 64, 2);
  pool_mask<<<divup((long)2 * 32 * 32, 256), 256, 0, stream>>>(
      occ2, occ3, 4, 64, 64, 2, 32, 32, 2);
  pool_mask<<<divup((long)2 * 32 * 32, 256), 256, 0, stream>>>(
      occ3, occ4, 2, 32, 32, 2, 32, 32, 1);

#define LAUNCH_CONV(CIN, COUT, STRIDE, NT, CLAST, IN, L, MK, OUT, inHp, inWp, \
                    Dout, Hout, Wout, outHp, outWp)                           \
  {                                                                           \
    long waves = (long)(Dout) * (Hout) * ((Wout) / (16 * (NT)));              \
    dim3 grid(divup(waves, 8), (COUT) / 16, 1);                               \
    conv3_wmma<CIN, COUT, STRIDE, NT, CLAST><<<grid, 256, 0, stream>>>(       \
        IN, W[L], SS[L], TT[L], MK, OUT, inHp, inWp, Dout, Hout, Wout, outHp, \
        outWp);                                                               \
  }

  // ---- stage 0: 32 -> 32,32,32  (16,256,256) -> (8,128,128)
  LAUNCH_CONV(32, 32, 2, 4, true, x0p, 0, occ1, s0A, 258, 258, 8, 128, 128, 130, 130);
  LAUNCH_CONV(32, 32, 1, 4, true, s0A, 1, occ1, s0B, 130, 130, 8, 128, 128, 130, 130);
  LAUNCH_CONV(32, 32, 1, 4, true, s0B, 2, occ1, s0A, 130, 130, 8, 128, 128, 130, 130);
  // ---- stage 1: 32 -> 64,64,64  (8,128,128) -> (4,64,64)
  LAUNCH_CONV(32, 64, 2, 4, true, s0A, 3, occ2, s1A, 130, 130, 4, 64, 64, 66, 66);
  LAUNCH_CONV(64, 64, 1, 4, true, s1A, 4, occ2, s1B, 66, 66, 4, 64, 64, 66, 66);
  LAUNCH_CONV(64, 64, 1, 4, true, s1B, 5, occ2, s1A, 66, 66, 4, 64, 64, 66, 66);
  // ---- stage 2: 64 -> 64,64,64  (4,64,64) -> (2,32,32)
  LAUNCH_CONV(64, 64, 2, 2, true, s1A, 6, occ3, s2A, 66, 66, 2, 32, 32, 34, 34);
  LAUNCH_CONV(64, 64, 1, 2, true, s2A, 7, occ3, s2B, 34, 34, 2, 32, 32, 34, 34);
  LAUNCH_CONV(64, 64, 1, 2, true, s2B, 8, occ3, s2A, 34, 34, 2, 32, 32, 34, 34);
  // ---- out: 64 -> 128 (2,32,32), NCDHW directly into d_out
  LAUNCH_CONV(64, 128, 1, 2, false, s2A, 9, occ4, out, 34, 34, 2, 32, 32, 0, 0);
#undef LAUNCH_CONV
}